// SoftRankIRLSStepModuleBase_2783138808066
// MI455X (gfx1250) — compile-verified
//
#include <hip/hip_runtime.h>

typedef __attribute__((ext_vector_type(2))) float v2f;
typedef __attribute__((ext_vector_type(8))) float v8f;

#define WAVES_PER_BLOCK 8
#define ROW_STRIDE 17              // floats; gcd(17,64)=1 -> conflict-free banks
#define SLICE (32 * ROW_STRIDE)    // 544 floats per wave

__global__ __launch_bounds__(32 * WAVES_PER_BLOCK)
void irls_eig_step_kernel(const float* __restrict__ G,
                          const float* __restrict__ F,
                          float* __restrict__ out,
                          int N, int Btot)
{
    __shared__ float lds[WAVES_PER_BLOCK * SLICE];
    const int lane = threadIdx.x & 31;
    const int wave = threadIdx.x >> 5;
    const int b = blockIdx.x * WAVES_PER_BLOCK + wave;
    if (b >= Btot) return;                    // wave-uniform guard, EXEC stays full
    float* S = lds + wave * SLICE;

    // Zero the whole slice once: columns 9..16 must stay 0 so WMMA rows 9..15 are 0.
    #pragma unroll
    for (int i = 0; i < ROW_STRIDE; ++i) S[i * 32 + lane] = 0.0f;

    // f estimate for this batch (broadcast, wave-uniform addresses)
    float fv[9];
    #pragma unroll
    for (int i = 0; i < 9; ++i) fv[i] = F[b * 9 + i];

    const float* __restrict__ Gb = G + (size_t)b * (size_t)N * 9;

    v8f acc = {};                              // 16x16 f32 Gram accumulator (rows/cols 0..8 live)
    const int mrow  = lane & 15;               // A/B fragment M (and N) index
    const int pbase = (lane >> 4) * 2;         // lanes 0-15 -> K=0,1 ; lanes 16-31 -> K=2,3
    const int chunks = N >> 5;                 // 32 points per chunk

    for (int c = 0; c < chunks; ++c) {
        const float* gp = Gb + (size_t)(c * 32 + lane) * 9;
        if (c + 1 < chunks) __builtin_prefetch(gp + 32 * 9, 0, 1);

        float g[9];
        #pragma unroll
        for (int i = 0; i < 9; ++i) g[i] = gp[i];

        // residual r = <g, f>, scale s = (r^2 + 1e-8)^(-1/4) so M = (sG)^T (sG)
        float r = 0.0f;
        #pragma unroll
        for (int i = 0; i < 9; ++i) r = fmaf(g[i], fv[i], r);
        float s = rsqrtf(sqrtf(fmaf(r, r, 1e-8f)));

        // weighted row into padded LDS tile
        #pragma unroll
        for (int i = 0; i < 9; ++i) S[lane * ROW_STRIDE + i] = g[i] * s;
        asm volatile("" ::: "memory");         // keep LDS stores before fragment loads

        // 8 x K=4 WMMA accumulations; A and B fragments are identical for a Gram product
        #pragma unroll
        for (int j0 = 0; j0 < 32; j0 += 4) {
            v2f a;
            a.x = S[(j0 + pbase + 0) * ROW_STRIDE + mrow];
            a.y = S[(j0 + pbase + 1) * ROW_STRIDE + mrow];
            acc = __builtin_amdgcn_wmma_f32_16x16x4_f32(
                false, a, false, a, (short)0, acc, false, false);
        }
        asm volatile("" ::: "memory");         // keep fragment loads before next chunk's stores
    }

    // Dump D to LDS with +1e-6 on the diagonal.
    // D layout: lane L<16 -> (M=v, N=L); lane L>=16 -> (M=v+8, N=L-16), v = VGPR 0..7.
    const int Ncol = lane & 15;
    const int mhi  = (lane >> 4) * 8;
    #pragma unroll
    for (int v = 0; v < 8; ++v) {
        int M = v + mhi;
        float val = acc[v];
        if (M == Ncol && M < 9) val += 1e-6f;
        S[M * ROW_STRIDE + Ncol] = val;
    }
    asm volatile("" ::: "memory");

    // Lane k (k<9) owns row k of M.
    float Mr[9];
    #pragma unroll
    for (int j = 0; j < 9; ++j) {
        float t = S[lane * ROW_STRIDE + j];    // in-bounds for all lanes (31*17+8 < 544)
        Mr[j] = (lane < 9) ? t : 0.0f;
    }

    // Gershgorin upper bound -> shift so that (shift*I - M) power-iterates to min eigvec
    float rs = 0.0f;
    #pragma unroll
    for (int j = 0; j < 9; ++j) rs += fabsf(Mr[j]);
    #pragma unroll
    for (int m = 16; m >= 1; m >>= 1) rs = fmaxf(rs, __shfl_xor(rs, m, 32));
    const float shift = rs * 1.0000001f + 1e-6f;

    // x0 = f (previous IRLS iterate: excellent seed for the smallest eigenvector)
    float y = 0.0f;
    #pragma unroll
    for (int j = 0; j < 9; ++j) if (lane == j) y = fv[j];

    for (int it = 0; it < 64; ++it) {
        float yj[9];
        #pragma unroll
        for (int j = 0; j < 9; ++j) yj[j] = __shfl(y, j, 32);
        float n2 = 1e-30f;
        #pragma unroll
        for (int j = 0; j < 9; ++j) n2 = fmaf(yj[j], yj[j], n2);
        const float inv = rsqrtf(n2);
        float t = 0.0f;
        #pragma unroll
        for (int j = 0; j < 9; ++j) t = fmaf(Mr[j], yj[j], t);
        y = inv * fmaf(shift, y, -t);          // y <- (shift*x - M x), x = y*inv
    }

    // final normalization + store
    {
        float n2 = 1e-30f;
        #pragma unroll
        for (int j = 0; j < 9; ++j) {
            float v = __shfl(y, j, 32);
            n2 = fmaf(v, v, n2);
        }
        y *= rsqrtf(n2);
    }
    if (lane < 9) out[b * 9 + lane] = y;
}

extern "C" void kernel_launch(void* const* d_in, const int* in_sizes, int n_in,
                              void* d_out, int out_size, void* d_ws, size_t ws_size,
                              hipStream_t stream) {
    const float* G = (const float*)d_in[0];   // [B, N, 9] fp32
    const float* F = (const float*)d_in[1];   // [B, 9]    fp32
    float* out = (float*)d_out;               // [B, 9]    fp32

    const int B = in_sizes[1] / 9;
    const int N = (int)((long long)in_sizes[0] / ((long long)B * 9)); // 2048

    const int blocks = (B + WAVES_PER_BLOCK - 1) / WAVES_PER_BLOCK;
    irls_eig_step_kernel<<<blocks, 32 * WAVES_PER_BLOCK, 0, stream>>>(G, F, out, N, B);
}